// SA_Module_65764539236949
// MI455X (gfx1250) — compile-verified
//
#include <hip/hip_runtime.h>

// Self-attention (SAGAN-style) for B=32, C=512, H=W=32 on gfx1250.
// GEMMs on V_WMMA_F32_16X16X4_F32 (exact fp32 FMA, matches fp32 reference
// through the exp-sensitive softmax). Block-shared A-operand tiles are staged
// into LDS with GLOBAL_LOAD_ASYNC_TO_LDS_* (ASYNCcnt) and read back as
// bank-conflict-free ds_load_b64 fragments.
//
// d_out (floats): attention[32*1024*1024] | rgb[32*512*1024] | hha[32*512*1024]
// d_ws  (floats): q[32*64*1024] | k[32*64*1024] | v[32*512*1024]  (~80 MB)

typedef float v2f __attribute__((ext_vector_type(2)));
typedef float v8f __attribute__((ext_vector_type(8)));

#define B_N   32
#define C_N   512
#define C8_N  64
#define HW_N  1024

static __device__ __forceinline__ v8f wmma4(v2f a, v2f b, v8f c) {
  // D = A(16x4,f32) * B(4x16,f32) + C(16x16,f32)
  return __builtin_amdgcn_wmma_f32_16x16x4_f32(
      /*neg_a=*/false, a, /*neg_b=*/false, b,
      /*c_mod=*/(short)0, c, /*reuse_a=*/false, /*reuse_b=*/false);
}

// Low 32 bits of a generic pointer to __shared__ == LDS byte offset
// (local->flat addrspacecast keeps the offset in the low word on AMDGPU).
static __device__ __forceinline__ unsigned lds_off(const void* p) {
  return (unsigned)(size_t)p;
}

// Async global->LDS, 16 bytes per lane (ISA 10.x, tracked by ASYNCcnt).
static __device__ __forceinline__ void async_b128(unsigned lds, const float* g) {
  asm volatile("global_load_async_to_lds_b128 %0, %1, off"
               :: "v"(lds), "v"(g) : "memory");
}
// Async global->LDS, 4 bytes per lane (lets us transpose while staging).
static __device__ __forceinline__ void async_b32(unsigned lds, const float* g) {
  asm volatile("global_load_async_to_lds_b32 %0, %1, off"
               :: "v"(lds), "v"(g) : "memory");
}
static __device__ __forceinline__ void wait_async0() {
#if __has_builtin(__builtin_amdgcn_s_wait_asynccnt)
  __builtin_amdgcn_s_wait_asynccnt(0);
#else
  asm volatile("s_wait_asynccnt 0" ::: "memory");
#endif
}

// ---------------------------------------------------------------------------
// K1: fused QKV projection.  Y[o,p] = sum_c W[o,c]*X[c,p] + bias[o]
// W rows 0..63 -> q, 64..127 -> k, 128..639 -> v.
// All 8 waves of a block share (b, mtile): stage the 16x512 W tile in LDS.
// Row stride 516 floats (516 % 64 == 4) -> the half-wave ds_load_b64 A-frag
// pattern (banks 4*l16 + k + kk0 + {0,1}) covers all 64 banks exactly once.
// ---------------------------------------------------------------------------
__global__ __launch_bounds__(256) void qkv_gemm_kernel(
    const float* __restrict__ x,
    const float* __restrict__ wq, const float* __restrict__ bq,
    const float* __restrict__ wk, const float* __restrict__ bk,
    const float* __restrict__ wv, const float* __restrict__ bv,
    float* __restrict__ qo, float* __restrict__ ko, float* __restrict__ vo)
{
  __shared__ float wtile[16 * 516];            // 33 KB

  const int tid  = threadIdx.x;
  const int wid  = blockIdx.x * 8 + (tid >> 5);
  const int ng   = wid & 15;          // per-wave 64-wide N group
  const int tmp  = wid >> 4;          // uniform across the block
  const int mt   = tmp % 40;
  const int b    = tmp / 40;
  const int lane = tid & 31;
  const int l16  = lane & 15;
  const int hi8  = (lane >> 4) << 3;  // C/D row offset for upper half-wave
  const int kk0  = (lane >> 4) << 1;  // K offset for upper half-wave

  const int orow0 = mt * 16;
  const float* W; const float* bias; float* out; int m0;
  if (orow0 < 64)       { W = wq; bias = bq; out = qo + (size_t)b * C8_N * HW_N; m0 = orow0; }
  else if (orow0 < 128) { W = wk; bias = bk; out = ko + (size_t)b * C8_N * HW_N; m0 = orow0 - 64; }
  else                  { W = wv; bias = bv; out = vo + (size_t)b * C_N  * HW_N; m0 = orow0 - 128; }

  // Stage W tile: 16 rows x 512 = 2048 float4, 8 per thread.
  #pragma unroll
  for (int i = 0; i < 8; ++i) {
    const int f   = tid + i * 256;     // float4 index
    const int row = f >> 7;            // 128 float4 per row
    const int c4  = f & 127;
    async_b128(lds_off(&wtile[row * 516 + c4 * 4]),
               W + (size_t)(m0 + row) * C_N + c4 * 4);
  }
  wait_async0();
  __syncthreads();

  const float* X  = x + (size_t)b * C_N * HW_N;
  const int    p0 = ng * 64;

  v8f acc[4];
  #pragma unroll
  for (int r = 0; r < 8; ++r) {
    const float bb = bias[m0 + hi8 + r];
    acc[0][r] = bb; acc[1][r] = bb; acc[2][r] = bb; acc[3][r] = bb;
  }

  const float* xcol = X + p0 + l16;
  const float* arow = &wtile[l16 * 516 + kk0];
  #pragma unroll 4
  for (int k = 0; k < C_N; k += 4) {
    v2f a = *(const v2f*)(arow + k);                       // ds_load_b64
    const float* xc = xcol + (size_t)(k + kk0) * HW_N;
    #pragma unroll
    for (int s = 0; s < 4; ++s) {                          // A reused 4x
      v2f bb;
      bb.x = xc[s * 16];
      bb.y = xc[s * 16 + HW_N];
      acc[s] = wmma4(a, bb, acc[s]);
    }
  }

  #pragma unroll
  for (int s = 0; s < 4; ++s)
    #pragma unroll
    for (int r = 0; r < 8; ++r)
      out[(size_t)(m0 + hi8 + r) * HW_N + p0 + s * 16 + l16] = acc[s][r];
}

// ---------------------------------------------------------------------------
// K2: energy = q^T k.  E[i,j] = sum_{c<64} q[c,i]*k[c,j].
// Stage the block-shared 16x64 q^T tile (transposed while staging via b32
// async copies), stride 68 (% 64 == 4) -> conflict-free A-frags.
// ---------------------------------------------------------------------------
__global__ __launch_bounds__(256) void energy_kernel(
    const float* __restrict__ q, const float* __restrict__ k,
    float* __restrict__ attn)
{
  __shared__ float qtile[16 * 68];             // 4.3 KB, layout [m][kk]

  const int tid  = threadIdx.x;
  const int wid  = blockIdx.x * 8 + (tid >> 5);
  const int ng   = wid & 15;
  const int tmp  = wid >> 4;                   // uniform across the block
  const int mt   = tmp & 63;
  const int b    = tmp >> 6;
  const int lane = tid & 31;
  const int l16  = lane & 15;
  const int hi8  = (lane >> 4) << 3;
  const int kk0  = (lane >> 4) << 1;

  const float* Q = q    + (size_t)b * C8_N * HW_N;
  const float* K = k    + (size_t)b * C8_N * HW_N;
  float*       E = attn + (size_t)b * HW_N * HW_N;
  const int i0 = mt * 16;
  const int j0 = ng * 64;

  // Stage q^T tile: element (m, kk) <- Q[kk][i0+m]; 1024 elems, 4 per thread.
  #pragma unroll
  for (int i = 0; i < 4; ++i) {
    const int e  = tid + i * 256;
    const int kk = e >> 4;
    const int m  = e & 15;
    async_b32(lds_off(&qtile[m * 68 + kk]),
              Q + (size_t)kk * HW_N + i0 + m);
  }
  wait_async0();
  __syncthreads();

  v8f acc[4];
  #pragma unroll
  for (int s = 0; s < 4; ++s)
    #pragma unroll
    for (int r = 0; r < 8; ++r) acc[s][r] = 0.0f;

  const float* arow = &qtile[l16 * 68 + kk0];
  #pragma unroll 4
  for (int kk = 0; kk < C8_N; kk += 4) {
    v2f a = *(const v2f*)(arow + kk);                      // ds_load_b64
    const float* kp = K + (size_t)(kk + kk0) * HW_N + j0 + l16;
    #pragma unroll
    for (int s = 0; s < 4; ++s) {
      v2f bb;
      bb.x = kp[s * 16];
      bb.y = kp[s * 16 + HW_N];
      acc[s] = wmma4(a, bb, acc[s]);
    }
  }

  #pragma unroll
  for (int s = 0; s < 4; ++s)
    #pragma unroll
    for (int r = 0; r < 8; ++r)
      E[(size_t)(i0 + hi8 + r) * HW_N + j0 + s * 16 + l16] = acc[s][r];
}

// ---------------------------------------------------------------------------
// K3: in-place row softmax over 1024 columns. One 256-thread block per row.
// ---------------------------------------------------------------------------
__global__ __launch_bounds__(256) void softmax_kernel(float* __restrict__ attn)
{
  const int tid = threadIdx.x;
  float* p = attn + (size_t)blockIdx.x * HW_N;
  float4 v = reinterpret_cast<float4*>(p)[tid];

  __shared__ float red[256];
  red[tid] = fmaxf(fmaxf(v.x, v.y), fmaxf(v.z, v.w));
  __syncthreads();
  for (int s = 128; s > 0; s >>= 1) {
    if (tid < s) red[tid] = fmaxf(red[tid], red[tid + s]);
    __syncthreads();
  }
  const float rmax = red[0];
  __syncthreads();

  v.x = __expf(v.x - rmax);
  v.y = __expf(v.y - rmax);
  v.z = __expf(v.z - rmax);
  v.w = __expf(v.w - rmax);

  red[tid] = v.x + v.y + v.z + v.w;
  __syncthreads();
  for (int s = 128; s > 0; s >>= 1) {
    if (tid < s) red[tid] += red[tid + s];
    __syncthreads();
  }
  const float inv = 1.0f / red[0];

  v.x *= inv; v.y *= inv; v.z *= inv; v.w *= inv;
  reinterpret_cast<float4*>(p)[tid] = v;
}

// ---------------------------------------------------------------------------
// K4: out[c,i] = sum_j v[c,j]*attn[i,j]; fused gamma*out + x / + hha.
// Dominant GEMM (34 GFLOP). Block-shared v rows staged in 16x256 chunks,
// double-buffered async (issue chunk c+1 while WMMAs eat chunk c).
// Stride 260 (% 64 == 4) -> conflict-free ds_load_b64 A-frags.
// B-operand = attn rows: contiguous float2 per lane (the transpose falls out
// of the 16x16x4 half-wave K layout), served by L2 (4 MB/batch, 192 MB L2).
// ---------------------------------------------------------------------------
#define K4_CHUNK 256

__global__ __launch_bounds__(256) void out_kernel(
    const float* __restrict__ v, const float* __restrict__ attn,
    const float* __restrict__ x, const float* __restrict__ hha,
    const float* __restrict__ gamma,
    float* __restrict__ rgb, float* __restrict__ hho)
{
  __shared__ float vbuf[2][16 * 260];          // 2 x 16.6 KB

  const int tid  = threadIdx.x;
  const int wid  = blockIdx.x * 8 + (tid >> 5);
  const int ng   = wid & 15;
  const int tmp  = wid >> 4;                   // uniform across the block
  const int mt   = tmp & 31;
  const int b    = tmp >> 5;
  const int lane = tid & 31;
  const int l16  = lane & 15;
  const int hi8  = (lane >> 4) << 3;
  const int kk0  = (lane >> 4) << 1;

  const float* V = v    + (size_t)b * C_N  * HW_N;
  const float* A = attn + (size_t)b * HW_N * HW_N;
  const int c0 = mt * 16;
  const int i0 = ng * 64;

  // Stage one 16x256 chunk of V rows: 1024 float4, 4 per thread.
  auto stage = [&](int ck, int bi) {
    #pragma unroll
    for (int i = 0; i < 4; ++i) {
      const int f   = tid + i * 256;   // float4 index
      const int row = f >> 6;          // 64 float4 per row
      const int c4  = f & 63;
      async_b128(lds_off(&vbuf[bi][row * 260 + c4 * 4]),
                 V + (size_t)(c0 + row) * HW_N + ck * K4_CHUNK + c4 * 4);
    }
  };

  v8f acc[4];
  #pragma unroll
  for (int s = 0; s < 4; ++s)
    #pragma unroll
    for (int r = 0; r < 8; ++r) acc[s][r] = 0.0f;

  stage(0, 0);
  wait_async0();
  __syncthreads();

  for (int ck = 0; ck < HW_N / K4_CHUNK; ++ck) {
    if (ck + 1 < HW_N / K4_CHUNK) stage(ck + 1, (ck + 1) & 1);

    const float* arow = &vbuf[ck & 1][l16 * 260 + kk0];
    const int    kb   = ck * K4_CHUNK;
    #pragma unroll 4
    for (int k = 0; k < K4_CHUNK; k += 4) {
      v2f a = *(const v2f*)(arow + k);                     // ds_load_b64
      #pragma unroll
      for (int s = 0; s < 4; ++s) {
        v2f bb = *(const v2f*)(A + (size_t)(i0 + s * 16 + l16) * HW_N
                                 + kb + k + kk0);
        acc[s] = wmma4(a, bb, acc[s]);
      }
    }

    wait_async0();      // next chunk landed
    __syncthreads();    // everyone done reading current buffer
  }

  const float g = gamma[0];
  const size_t base = (size_t)b * C_N * HW_N;
  #pragma unroll
  for (int s = 0; s < 4; ++s)
    #pragma unroll
    for (int r = 0; r < 8; ++r) {
      const size_t idx = base + (size_t)(c0 + hi8 + r) * HW_N + i0 + s * 16 + l16;
      const float o = g * acc[s][r];
      rgb[idx] = o + x[idx];
      hho[idx] = o + hha[idx];
    }
}

// ---------------------------------------------------------------------------
extern "C" void kernel_launch(void* const* d_in, const int* in_sizes, int n_in,
                              void* d_out, int out_size, void* d_ws, size_t ws_size,
                              hipStream_t stream)
{
  (void)in_sizes; (void)n_in; (void)out_size; (void)ws_size;

  const float* x     = (const float*)d_in[0];
  const float* hha   = (const float*)d_in[1];
  const float* wq    = (const float*)d_in[2];
  const float* bq    = (const float*)d_in[3];
  const float* wk    = (const float*)d_in[4];
  const float* bk    = (const float*)d_in[5];
  const float* wv    = (const float*)d_in[6];
  const float* bv    = (const float*)d_in[7];
  const float* gamma = (const float*)d_in[8];

  float* attn = (float*)d_out;
  float* rgb  = attn + (size_t)B_N * HW_N * HW_N;
  float* hho  = rgb  + (size_t)B_N * C_N  * HW_N;

  float* q = (float*)d_ws;                       //  8 MB
  float* k = q + (size_t)B_N * C8_N * HW_N;      //  8 MB
  float* v = k + (size_t)B_N * C8_N * HW_N;      // 64 MB

  // waves: B*40*16 = 20480 -> 2560 blocks of 8 waves
  qkv_gemm_kernel<<<B_N * 40 * 16 / 8, 256, 0, stream>>>(
      x, wq, bq, wk, bk, wv, bv, q, k, v);
  // waves: B*64*16 = 32768 -> 4096 blocks
  energy_kernel<<<B_N * 64 * 16 / 8, 256, 0, stream>>>(q, k, attn);
  // one block per attention row: 32768 blocks
  softmax_kernel<<<B_N * HW_N, 256, 0, stream>>>(attn);
  // waves: B*32*16 = 16384 -> 2048 blocks
  out_kernel<<<B_N * 32 * 16 / 8, 256, 0, stream>>>(
      v, attn, x, hha, gamma, rgb, hho);
}